// EQSS2D_37726992728131
// MI455X (gfx1250) — compile-verified
//
#include <hip/hip_runtime.h>
#include <math.h>

// ---------------------------------------------------------------------------
// SS2D (VMamba selective scan) for MI455X / gfx1250, wave32.
// B=2, K=4, D=96, H=W=48 -> L=2304, R=6, N=16.
//
// Pipeline:
//  1) k_fuse_w : Wf[k] (128x96) = [dt_proj_w @ x_proj_w[:R] ; x_proj_w[R:]]
//  2) k_gather : cross-scan permutation xs -> xs_seq, stored L-MAJOR [bk][l][d]
//  3) k_gemm   : proj[bk][l][row] = Wf[k] @ xs_seq  via v_wmma_f32_16x16x4_f32
//                (b64 A/B fragment loads, 2x b128 accumulator stores)
//  4) k_phase1 : chunked scan pass 1 (local scans, h0=0) -> h_end, sum(dt)
//  5) k_phase2 : sequential carry propagation (chunk decay = exp(A*sum_dt))
//  6) k_phase3 : rescan with carried h0, y = h.C + Ds*x, scatter cross-merge
// ---------------------------------------------------------------------------

#define BB   2
#define KK   4
#define DD   96
#define HH   48
#define WV   48
#define LL   (HH * WV)      // 2304
#define RR   6
#define NN   16
#define CDIM 38             // R + 2N
#define MROWS 128           // 96 dt rows + 16 B rows + 16 C rows
#define LC   48             // chunk length
#define NCH  (LL / LC)      // 48 chunks

typedef __attribute__((ext_vector_type(2))) float v2f;
typedef __attribute__((ext_vector_type(4))) float v4f;
typedef __attribute__((ext_vector_type(8))) float v8f;

__device__ __forceinline__ float softplusf(float x) {
  // jax.nn.softplus: max(x,0) + log1p(exp(-|x|)) (numerically stable)
  return fmaxf(x, 0.0f) + log1pf(expf(-fabsf(x)));
}

// ---------------------------------------------------------------------------
// 1) Fused weights (row-major [k][r][d]):
//    rows 0..95   = dt_proj_w(96x6) @ x_proj_w[:6](6x96)
//    rows 96..127 = x_proj_w[6:38]  (B then C rows)
// ---------------------------------------------------------------------------
__global__ void k_fuse_w(const float* __restrict__ xproj,
                         const float* __restrict__ dtw,
                         float* __restrict__ Wf) {
  int t = blockIdx.x * blockDim.x + threadIdx.x;
  if (t >= KK * MROWS * DD) return;
  int d = t % DD;
  int r = (t / DD) % MROWS;
  int k = t / (DD * MROWS);
  const float* xp = xproj + (size_t)k * CDIM * DD;
  float v;
  if (r < DD) {
    float s = 0.0f;
#pragma unroll
    for (int j = 0; j < RR; ++j)
      s += dtw[((size_t)k * DD + r) * RR + j] * xp[j * DD + d];
    v = s;
  } else {
    v = xp[(r - DD + RR) * DD + d];
  }
  Wf[((size_t)k * MROWS + r) * DD + d] = v;
}

// ---------------------------------------------------------------------------
// 2) Cross-scan gather into L-MAJOR layout:
//    xs_seq[((bk)*LL + l)*DD + d] = xs[((bk)*DD + d)*LL + perm_k(l)]
//    (coalesced writes, scattered L2-resident reads)
// ---------------------------------------------------------------------------
__global__ void k_gather(const float* __restrict__ xs,
                         float* __restrict__ xseq) {
  int t = blockIdx.x * blockDim.x + threadIdx.x;
  if (t >= BB * KK * LL * DD) return;
  int d  = t % DD;
  int l  = (t / DD) % LL;
  int bk = t / (DD * LL);
  int k  = bk % KK;
  int src;
  if (k == 0)       src = l;
  else if (k == 1)  src = (HH - 1 - (l % HH)) * WV + (l / HH);   // rot90 k=-1
  else if (k == 2)  src = LL - 1 - l;                            // reversed
  else              src = (l % HH) * WV + (WV - 1 - (l / HH));   // rot90 k=+1
  xseq[t] = xs[((size_t)bk * DD + d) * LL + src];
}

// ---------------------------------------------------------------------------
// 3) Projection GEMM with fp32 WMMA (16x16x4).
//    One block = 8 waves; wave m computes M-tile m of a 16-wide L column tile.
//    A 16x4 frag: lane holds row (lane&15), K = kk*4 + (lane>>4)*2 + {0,1}
//    B 4x16 frag: lane holds col (lane&15), K = kk*4 + (lane>>4)*2 + {0,1}
//    C/D 16x16:   VGPR r -> row r + (lane>>4)*8, col = lane&15
//    L-major xseq makes B frags a single b64 load; L-major proj makes the
//    8 accumulator rows per lane contiguous -> two b128 stores.
// ---------------------------------------------------------------------------
__global__ void __launch_bounds__(256)
k_gemm(const float* __restrict__ Wf,
       const float* __restrict__ xseq,
       float* __restrict__ proj) {
  int bid = blockIdx.x;                 // B*K*(L/16) = 1152 blocks
  int lt  = bid % (LL / 16);
  int k   = (bid / (LL / 16)) % KK;
  int bk  = (bid / (LL / 16));          // b*KK + k
  int lane  = threadIdx.x & 31;
  int mtile = threadIdx.x >> 5;         // 0..7

  const float* Wk = Wf + (size_t)k * MROWS * DD;
  const float* Xt = xseq + (size_t)bk * LL * DD;

  const int mrow = mtile * 16 + (lane & 15);
  const int col  = lt * 16 + (lane & 15);
  const int khi  = (lane >> 4) * 2;

  v8f acc = {};
#pragma unroll
  for (int kk = 0; kk < DD / 4; ++kk) {
    int kb = kk * 4 + khi;              // even -> 8B aligned
    v2f a  = *(const v2f*)(Wk + (size_t)mrow * DD + kb);
    v2f bf = *(const v2f*)(Xt + (size_t)col * DD + kb);
    acc = __builtin_amdgcn_wmma_f32_16x16x4_f32(
        /*neg_a=*/false, a, /*neg_b=*/false, bf,
        /*c_mod=*/(short)0, acc, /*reuse_a=*/false, /*reuse_b=*/false);
  }

  float* P = proj + ((size_t)bk * LL + col) * MROWS
                  + mtile * 16 + (lane >> 4) * 8;   // 32B aligned
  v4f lo = {acc[0], acc[1], acc[2], acc[3]};
  v4f hi = {acc[4], acc[5], acc[6], acc[7]};
  *(v4f*)(P)     = lo;
  *(v4f*)(P + 4) = hi;
}

// ---------------------------------------------------------------------------
// 4) Phase 1: local chunk scans with h0 = 0. One block per (b,k,chunk),
//    96 lanes, lane = d. B rows staged in LDS as [i][n] (broadcast reads).
//    dt/x reads are coalesced across lanes each step (L-major layouts).
// ---------------------------------------------------------------------------
__global__ void __launch_bounds__(DD)
k_phase1(const float* __restrict__ proj, const float* __restrict__ xseq,
         const float* __restrict__ bias, const float* __restrict__ A_log,
         float* __restrict__ hend, float* __restrict__ Ssum) {
  int bid = blockIdx.x;
  int c  = bid % NCH;
  int bk = bid / NCH;
  int k  = bk % KK;
  int d  = threadIdx.x;

  const float* pb = proj + (size_t)bk * LL * MROWS;
  const float* xb = xseq + (size_t)bk * LL * DD;

  __shared__ float sB[LC * NN];         // [i][n]
  for (int idx = threadIdx.x; idx < LC * NN; idx += blockDim.x) {
    int i = idx / NN, n = idx % NN;
    sB[idx] = pb[(size_t)(c * LC + i) * MROWS + DD + n];
  }
  __syncthreads();

  float A[NN], h[NN];
#pragma unroll
  for (int n = 0; n < NN; ++n) {
    A[n] = -expf(A_log[((size_t)k * DD + d) * NN + n]);
    h[n] = 0.0f;
  }
  float S = 0.0f;
  const float bsv = bias[k * DD + d];

  for (int i = 0; i < LC; ++i) {
    int l = c * LC + i;
    float dt = softplusf(pb[(size_t)l * MROWS + d] + bsv);
    float x  = xb[(size_t)l * DD + d];
    S += dt;
    float dtx = dt * x;
#pragma unroll
    for (int n = 0; n < NN; ++n)
      h[n] = expf(dt * A[n]) * h[n] + dtx * sB[i * NN + n];
  }

  size_t hbase = ((size_t)(bk * NCH + c) * DD + d) * NN;
#pragma unroll
  for (int n = 0; n < NN; ++n) hend[hbase + n] = h[n];
  Ssum[(size_t)(bk * NCH + c) * DD + d] = S;
}

// ---------------------------------------------------------------------------
// 5) Phase 2: sequential carry across chunks. Chunk decay for state n is
//    prod_l exp(dt_l*A_n) = exp(A_n * sum_l dt_l). 12288 independent threads.
// ---------------------------------------------------------------------------
__global__ void k_phase2(const float* __restrict__ A_log,
                         const float* __restrict__ Ssum,
                         const float* __restrict__ hend,
                         float* __restrict__ hinit) {
  int t = blockIdx.x * blockDim.x + threadIdx.x;
  if (t >= BB * KK * DD * NN) return;
  int n  = t % NN;
  int d  = (t / NN) % DD;
  int bk = t / (NN * DD);
  int k  = bk % KK;
  float A = -expf(A_log[((size_t)k * DD + d) * NN + n]);
  float carry = 0.0f;
  for (int c = 0; c < NCH; ++c) {
    size_t base = ((size_t)(bk * NCH + c) * DD + d) * NN + n;
    hinit[base] = carry;
    float S = Ssum[(size_t)(bk * NCH + c) * DD + d];
    carry = expf(A * S) * carry + hend[base];
  }
}

// ---------------------------------------------------------------------------
// 6) Phase 3: rescan with carried h0; y = sum_n h_n*C_n + Ds*x; write through
//    the cross-merge (inverse) permutation straight into (B, D*K, H, W).
// ---------------------------------------------------------------------------
__global__ void __launch_bounds__(DD)
k_phase3(const float* __restrict__ proj, const float* __restrict__ xseq,
         const float* __restrict__ bias, const float* __restrict__ A_log,
         const float* __restrict__ Ds,  const float* __restrict__ hinit,
         float* __restrict__ out) {
  int bid = blockIdx.x;
  int c  = bid % NCH;
  int bk = bid / NCH;
  int k  = bk % KK;
  int b  = bk / KK;
  int d  = threadIdx.x;

  const float* pb = proj + (size_t)bk * LL * MROWS;
  const float* xb = xseq + (size_t)bk * LL * DD;

  __shared__ float sB[LC * NN];         // [i][n]
  __shared__ float sC[LC * NN];
  for (int idx = threadIdx.x; idx < LC * NN; idx += blockDim.x) {
    int i = idx / NN, n = idx % NN;
    sB[idx] = pb[(size_t)(c * LC + i) * MROWS + DD + n];
    sC[idx] = pb[(size_t)(c * LC + i) * MROWS + DD + NN + n];
  }
  __syncthreads();

  float A[NN], h[NN];
  size_t hbase = ((size_t)(bk * NCH + c) * DD + d) * NN;
#pragma unroll
  for (int n = 0; n < NN; ++n) {
    A[n] = -expf(A_log[((size_t)k * DD + d) * NN + n]);
    h[n] = hinit[hbase + n];
  }
  const float bsv = bias[k * DD + d];
  const float dsv = Ds[k * DD + d];

  for (int i = 0; i < LC; ++i) {
    int l = c * LC + i;
    float dt = softplusf(pb[(size_t)l * MROWS + d] + bsv);
    float x  = xb[(size_t)l * DD + d];
    float dtx = dt * x;
    float y = 0.0f;
#pragma unroll
    for (int n = 0; n < NN; ++n) {
      h[n] = expf(dt * A[n]) * h[n] + dtx * sB[i * NN + n];
      y += h[n] * sC[i * NN + n];
    }
    float ov = y + dsv * x;

    int hh, ww;
    if (k == 0)      { hh = l / WV;                 ww = l % WV; }
    else if (k == 1) { hh = HH - 1 - (l % HH);      ww = l / HH; }
    else if (k == 2) { int lr = LL - 1 - l; hh = lr / WV; ww = lr % WV; }
    else             { hh = l % HH;                 ww = WV - 1 - (l / HH); }
    out[(((size_t)b * (DD * KK) + (size_t)d * KK + k) * HH + hh) * WV + ww] = ov;
  }
}

// ---------------------------------------------------------------------------
extern "C" void kernel_launch(void* const* d_in, const int* in_sizes, int n_in,
                              void* d_out, int out_size, void* d_ws, size_t ws_size,
                              hipStream_t stream) {
  const float* xs    = (const float*)d_in[0];
  const float* xproj = (const float*)d_in[1];
  const float* dtw   = (const float*)d_in[2];
  const float* bias  = (const float*)d_in[3];
  const float* A_log = (const float*)d_in[4];
  const float* Dsk   = (const float*)d_in[5];
  float* out = (float*)d_out;

  float* ws = (float*)d_ws;
  size_t off = 0;
  float* xseq  = ws + off; off += (size_t)BB * KK * LL * DD;       // 1,769,472
  float* Wf    = ws + off; off += (size_t)KK * MROWS * DD;         //    49,152
  float* proj  = ws + off; off += (size_t)BB * KK * LL * MROWS;    // 2,359,296
  float* hend  = ws + off; off += (size_t)BB * KK * NCH * DD * NN; //   589,824
  float* Ssum  = ws + off; off += (size_t)BB * KK * NCH * DD;      //    36,864
  float* hinit = ws + off; off += (size_t)BB * KK * NCH * DD * NN; //   589,824
  (void)ws_size; (void)in_sizes; (void)n_in; (void)out_size;       // ~21.6 MB total

  k_fuse_w<<<(KK * MROWS * DD + 255) / 256, 256, 0, stream>>>(xproj, dtw, Wf);
  k_gather<<<(BB * KK * LL * DD + 255) / 256, 256, 0, stream>>>(xs, xseq);
  k_gemm<<<BB * KK * (LL / 16), 256, 0, stream>>>(Wf, xseq, proj);
  k_phase1<<<BB * KK * NCH, DD, 0, stream>>>(proj, xseq, bias, A_log, hend, Ssum);
  k_phase2<<<(BB * KK * DD * NN + 255) / 256, 256, 0, stream>>>(A_log, Ssum, hend, hinit);
  k_phase3<<<BB * KK * NCH, DD, 0, stream>>>(proj, xseq, bias, A_log, Dsk, hinit, out);
}